// GRUInferenceLayer_52123723104482
// MI455X (gfx1250) — compile-verified
//
#include <hip/hip_runtime.h>
#include <math.h>

typedef __attribute__((ext_vector_type(2))) float v2f;
typedef __attribute__((ext_vector_type(4))) float v4f;
typedef __attribute__((ext_vector_type(8))) float v8f;

#define B_ 256
#define L_ 200
#define C_ 4
#define H_ 128
#define G3_ 384
#define XS 132   // padded LDS row stride for x/h tiles (floats)
#define PS 385   // padded LDS row stride for pre-activations
#define NTHREADS 768

// ---------------------------------------------------------------- lengths ----
__global__ void gru_lengths_kernel(const unsigned char* __restrict__ mask,
                                   int* __restrict__ lens) {
  int b = blockIdx.x * blockDim.x + threadIdx.x;
  if (b >= B_) return;
  const unsigned char* m = mask + (size_t)b * L_;
  int cnt = 0;
  for (int t = 0; t < L_; ++t) cnt += (m[t] != 0);
  lens[b] = L_ - cnt;
}

// ------------------------------------------- phase 1: xp = x @ W_ih^T + b ----
// Fully parallel WMMA GEMM on the UNROLLED seqs (the roll is a pure time-index
// permutation; the scan applies it by reading xp at tt=(t-len)%L).
__launch_bounds__(NTHREADS, 1)
__global__ void gru_xp_kernel(const float* __restrict__ seqs,
                              const float* __restrict__ W_ih,
                              const float* __restrict__ b_ih,
                              float* __restrict__ xp) {
  __shared__ float xl[16 * XS];

  const int tid  = threadIdx.x;
  const int lane = tid & 31;
  const int wave = tid >> 5;
  const int hi   = lane >> 4;
  const int lo   = lane & 15;
  const int cb   = blockIdx.x >> 3;       // 0..63
  const int c    = cb >> 4;
  const int b0   = (cb & 15) * 16;
  const int t0   = (blockIdx.x & 7) * 25; // 8 chunks x 25 steps
  const int gbase = wave * 16;

  v2f wih[32];
  {
    const float* wi = W_ih + ((size_t)c * G3_ + gbase + lo) * H_;
#pragma unroll
    for (int kc = 0; kc < 32; ++kc) wih[kc] = *(const v2f*)(wi + kc * 4 + 2 * hi);
  }
  const float bi = b_ih[(size_t)c * G3_ + gbase + lo];

  for (int s = 0; s < 25; ++s) {
    const int t = t0 + s;
    if (tid < 512) {
      const int row = tid >> 5;
      const int col = (tid & 31) * 4;
      const float* src = seqs + (((size_t)(b0 + row) * L_ + t) * C_ + c) * H_ + col;
      *(v4f*)&xl[row * XS + col] = *(const v4f*)src;
    }
    __syncthreads();
    v8f acc;
#pragma unroll
    for (int i = 0; i < 8; ++i) acc[i] = bi;
#pragma unroll
    for (int kc = 0; kc < 32; ++kc) {
      const int k = kc * 4 + 2 * hi;
      v2f a = *(const v2f*)&xl[lo * XS + k];
      acc = __builtin_amdgcn_wmma_f32_16x16x4_f32(false, a, false, wih[kc],
                                                  (short)0, acc, false, false);
    }
#pragma unroll
    for (int i = 0; i < 8; ++i) {
      const int row = i + 8 * hi;
      xp[(((size_t)c * B_ + b0 + row) * L_ + t) * G3_ + gbase + lo] = acc[i];
    }
    __syncthreads();
  }
}

// --------------------------------- phase 2: sequential scan (h-GEMM only) ----
// whh resident (64 VGPRs). All 32 A-chunks preloaded first (one DS wait), then
// 32 back-to-back accumulator-chained WMMAs. xp gathers issued before the GEMM
// so L2 latency hides behind it.
__launch_bounds__(NTHREADS, 1)
__global__ void gru_scan_xp_kernel(const float* __restrict__ xp,
                                   const float* __restrict__ W_hh,
                                   const float* __restrict__ b_hh,
                                   const int* __restrict__ lens,
                                   float* __restrict__ out) {
  __shared__ float hbuf[2][16 * XS];
  __shared__ float pre_h[16 * PS];
  __shared__ int   lens_s[16];

  const int tid  = threadIdx.x;
  const int lane = tid & 31;
  const int wave = tid >> 5;
  const int hi   = lane >> 4;
  const int lo   = lane & 15;
  const int c    = blockIdx.x >> 4;
  const int b0   = (blockIdx.x & 15) * 16;
  const int gbase = wave * 16;

  v2f whh[32];
  {
    const float* wh = W_hh + ((size_t)c * G3_ + gbase + lo) * H_;
#pragma unroll
    for (int kc = 0; kc < 32; ++kc) whh[kc] = *(const v2f*)(wh + kc * 4 + 2 * hi);
  }
  const float bh = b_hh[(size_t)c * G3_ + gbase + lo];

  if (tid < 16) lens_s[tid] = lens[b0 + tid];
  for (int i = tid; i < 16 * XS; i += NTHREADS) { hbuf[0][i] = 0.0f; hbuf[1][i] = 0.0f; }
  __syncthreads();

  const int nel = (tid < 512) ? 3 : 2;  // up to 3 (row,j) gate elements / thread

  int cur = 0;
  for (int t = 0; t < L_; ++t) {
    const float* hc = hbuf[cur];

    // issue xp gathers early (independent of h, hidden behind the WMMA chain)
    float xr[3], xz[3], xn[3];
    for (int e = 0; e < nel; ++e) {
      const int idx = tid + e * NTHREADS;
      const int row = (idx >> 7) & 15, j = idx & 127;
      int tt = t - lens_s[row]; if (tt < 0) tt += L_;
      const float* p = xp + (((size_t)c * B_ + b0 + row) * L_ + tt) * G3_;
      xr[e] = p[j];
      xz[e] = p[128 + j];
      xn[e] = p[256 + j];
    }

    // h-GEMM: preload all A chunks, then back-to-back chained WMMAs
    v2f a[32];
#pragma unroll
    for (int kc = 0; kc < 32; ++kc)
      a[kc] = *(const v2f*)&hc[lo * XS + kc * 4 + 2 * hi];

    v8f ah;
#pragma unroll
    for (int i = 0; i < 8; ++i) ah[i] = bh;
#pragma unroll
    for (int kc = 0; kc < 32; ++kc)
      ah = __builtin_amdgcn_wmma_f32_16x16x4_f32(false, a[kc], false, whh[kc],
                                                 (short)0, ah, false, false);

#pragma unroll
    for (int i = 0; i < 8; ++i) pre_h[(i + 8 * hi) * PS + gbase + lo] = ah[i];
    __syncthreads();

    float* hn_ptr = hbuf[cur ^ 1];
    for (int e = 0; e < nel; ++e) {
      const int idx = tid + e * NTHREADS;
      const int row = (idx >> 7) & 15, j = idx & 127;
      const float hr   = pre_h[row * PS + j];
      const float hz   = pre_h[row * PS + 128 + j];
      const float hnp  = pre_h[row * PS + 256 + j];
      const float hold = hc[row * XS + j];
      const float r = 1.0f / (1.0f + __expf(-(xr[e] + hr)));
      const float z = 1.0f / (1.0f + __expf(-(xz[e] + hz)));
      const float n = tanhf(xn[e] + r * hnp);
      const float hv = (1.0f - z) * n + z * hold;
      hn_ptr[row * XS + j] = hv;
      int tp = t - lens_s[row]; if (tp < 0) tp += L_;  // output index == tt
      out[(((size_t)c * B_ + b0 + row) * L_ + tp) * H_ + j] =
          (tp < lens_s[row]) ? hv : 0.0f;
    }
    __syncthreads();
    cur ^= 1;
  }
}

// ------------------------- fallback: fused scan with streamed W_ih (no ws) ---
__launch_bounds__(NTHREADS, 1)
__global__ void gru_fused_kernel(const float* __restrict__ seqs,
                                 const float* __restrict__ W_ih,
                                 const float* __restrict__ W_hh,
                                 const float* __restrict__ b_ih,
                                 const float* __restrict__ b_hh,
                                 const int* __restrict__ lens,
                                 float* __restrict__ out) {
  __shared__ float xlds[16 * XS];
  __shared__ float hbuf[2][16 * XS];
  __shared__ float pre_x[16 * PS];
  __shared__ float pre_hn[16 * XS];
  __shared__ int   lens_s[16];

  const int tid  = threadIdx.x;
  const int lane = tid & 31;
  const int wave = tid >> 5;
  const int hi   = lane >> 4;
  const int lo   = lane & 15;
  const int c    = blockIdx.x >> 4;
  const int b0   = (blockIdx.x & 15) * 16;
  const int gbase = wave * 16;

  v2f whh[32];  // only W_hh resident; W_ih streamed from L2 each step
  const float* wi = W_ih + ((size_t)c * G3_ + gbase + lo) * H_;
  {
    const float* wh = W_hh + ((size_t)c * G3_ + gbase + lo) * H_;
#pragma unroll
    for (int kc = 0; kc < 32; ++kc) whh[kc] = *(const v2f*)(wh + kc * 4 + 2 * hi);
  }
  const float bi = b_ih[(size_t)c * G3_ + gbase + lo];
  const float bh = b_hh[(size_t)c * G3_ + gbase + lo];

  if (tid < 16) lens_s[tid] = lens[b0 + tid];
  for (int i = tid; i < 16 * XS; i += NTHREADS) { hbuf[0][i] = 0.0f; hbuf[1][i] = 0.0f; }
  __syncthreads();

  if (tid < 512) {  // x_0
    const int row = tid >> 5, col = (tid & 31) * 4;
    int tt = -lens_s[row]; if (tt < 0) tt += L_;
    const float* src = seqs + (((size_t)(b0 + row) * L_ + tt) * C_ + c) * H_ + col;
    *(v4f*)&xlds[row * XS + col] = *(const v4f*)src;
  }
  __syncthreads();

  int cur = 0;
  for (int t = 0; t < L_; ++t) {
    const float* hc = hbuf[cur];
    v8f ax, ah;
#pragma unroll
    for (int i = 0; i < 8; ++i) { ax[i] = bi; ah[i] = bh; }

    v2f wbuf[2][8];
#pragma unroll
    for (int q = 0; q < 8; ++q) wbuf[0][q] = *(const v2f*)(wi + q * 4 + 2 * hi);

    // h-GEMM first: hides W_ih L2 latency behind the WMMA chain
#pragma unroll
    for (int kc = 0; kc < 32; ++kc) {
      const int k = kc * 4 + 2 * hi;
      v2f a = *(const v2f*)&hc[lo * XS + k];
      ah = __builtin_amdgcn_wmma_f32_16x16x4_f32(false, a, false, whh[kc],
                                                 (short)0, ah, false, false);
    }
    // x-GEMM in 4 groups of 8 K-chunks, double-buffered weight stream
#pragma unroll
    for (int g = 0; g < 4; ++g) {
      if (g < 3) {
#pragma unroll
        for (int q = 0; q < 8; ++q)
          wbuf[(g + 1) & 1][q] = *(const v2f*)(wi + (8 * (g + 1) + q) * 4 + 2 * hi);
      }
#pragma unroll
      for (int q = 0; q < 8; ++q) {
        const int k = (8 * g + q) * 4 + 2 * hi;
        v2f a = *(const v2f*)&xlds[lo * XS + k];
        ax = __builtin_amdgcn_wmma_f32_16x16x4_f32(false, a, false, wbuf[g & 1][q],
                                                   (short)0, ax, false, false);
      }
    }

    if (gbase < 256) {
#pragma unroll
      for (int i = 0; i < 8; ++i) pre_x[(i + 8 * hi) * PS + gbase + lo] = ax[i] + ah[i];
    } else {
#pragma unroll
      for (int i = 0; i < 8; ++i) {
        const int row = i + 8 * hi;
        pre_x[row * PS + gbase + lo] = ax[i];
        pre_hn[row * XS + (gbase - 256) + lo] = ah[i];
      }
    }
    __syncthreads();

    float* hn_ptr = hbuf[cur ^ 1];
    for (int idx = tid; idx < 2048; idx += NTHREADS) {
      const int row = idx >> 7, j = idx & 127;
      const float sr = pre_x[row * PS + j];
      const float sz = pre_x[row * PS + 128 + j];
      const float xin = pre_x[row * PS + 256 + j];
      const float hnp = pre_hn[row * XS + j];
      const float hold = hc[row * XS + j];
      const float r = 1.0f / (1.0f + __expf(-sr));
      const float z = 1.0f / (1.0f + __expf(-sz));
      const float n = tanhf(xin + r * hnp);
      const float hv = (1.0f - z) * n + z * hold;
      hn_ptr[row * XS + j] = hv;
      int tp = t - lens_s[row]; if (tp < 0) tp += L_;
      out[(((size_t)c * B_ + b0 + row) * L_ + tp) * H_ + j] = (tp < lens_s[row]) ? hv : 0.0f;
    }
    if (t + 1 < L_ && tid < 512) {
      const int row = tid >> 5, col = (tid & 31) * 4;
      int tt = (t + 1) - lens_s[row]; if (tt < 0) tt += L_;
      const float* src = seqs + (((size_t)(b0 + row) * L_ + tt) * C_ + c) * H_ + col;
      *(v4f*)&xlds[row * XS + col] = *(const v4f*)src;
    }
    __syncthreads();
    cur ^= 1;
  }
}

// ---------------------------------------------------------------- launch -----
extern "C" void kernel_launch(void* const* d_in, const int* in_sizes, int n_in,
                              void* d_out, int out_size, void* d_ws, size_t ws_size,
                              hipStream_t stream) {
  (void)in_sizes; (void)n_in; (void)out_size;
  const float*         seqs  = (const float*)d_in[0];
  const unsigned char* tmask = (const unsigned char*)d_in[1];
  const float* W_ih = (const float*)d_in[3];
  const float* W_hh = (const float*)d_in[4];
  const float* b_ih = (const float*)d_in[5];
  const float* b_hh = (const float*)d_in[6];
  float* out = (float*)d_out;

  int*   lens = (int*)d_ws;                              // 1 KB
  float* xp   = (float*)((char*)d_ws + 1024);            // C*B*L*3H fp32
  const size_t xp_bytes = (size_t)C_ * B_ * L_ * G3_ * sizeof(float);

  hipLaunchKernelGGL(gru_lengths_kernel, dim3(1), dim3(256), 0, stream, tmask, lens);

  if (ws_size >= 1024 + xp_bytes) {
    hipLaunchKernelGGL(gru_xp_kernel, dim3(512), dim3(NTHREADS), 0, stream,
                       seqs, W_ih, b_ih, xp);
    hipLaunchKernelGGL(gru_scan_xp_kernel, dim3(64), dim3(NTHREADS), 0, stream,
                       xp, W_hh, b_hh, lens, out);
  } else {
    hipLaunchKernelGGL(gru_fused_kernel, dim3(64), dim3(NTHREADS), 0, stream,
                       seqs, W_ih, W_hh, b_ih, b_hh, lens, out);
  }
}